// GCN_4698694222357
// MI455X (gfx1250) — compile-verified
//
#include <hip/hip_runtime.h>
#include <hip/hip_bf16.h>

typedef float v2f __attribute__((ext_vector_type(2)));
typedef float v8f __attribute__((ext_vector_type(8)));

#if __has_builtin(__builtin_amdgcn_wmma_f32_16x16x4_f32)
#define HAVE_WMMA_F32 1
#else
#define HAVE_WMMA_F32 0
#endif

constexpr int IN_F = 128;  // input features
constexpr int H_F  = 64;   // hidden features
constexpr int C_F  = 16;   // classes

// Native CDNA5 fp32 global atomic add, no-return form (STOREcnt-tracked).
// Guarantees GLOBAL_ATOMIC_ADD_F32 instead of a possible CAS-loop expansion.
__device__ __forceinline__ void atomic_add_f32_native(float* addr, float val)
{
    asm volatile("global_atomic_add_f32 %0, %1, off"
                 :: "v"(addr), "v"(val)
                 : "memory");
}

// ---------------------------------------------------------------------------
// GEMM1: T1[M,64] = X[M,128] @ W1[128,64]
// One wave per 16-row tile; 4 accumulators cover N=64.
// A 16x4 f32 fragment: lane&15 = M row; (lane>>4)*2 = K sub-offset (ISA 7.12.2)
// C/D: VGPR v -> row = v + 8*(lane>>4), col = lane&15 within the N-tile.
// ---------------------------------------------------------------------------
__global__ __launch_bounds__(256) void gcn_gemm1(
    const float* __restrict__ X, const float* __restrict__ W1,
    float* __restrict__ T1, int n_tiles)
{
    const int wave    = threadIdx.x >> 5;
    const int lane    = threadIdx.x & 31;
    const int tile    = blockIdx.x * 8 + wave;      // wave-uniform guard
    if (tile >= n_tiles) return;                    // whole wave exits: EXEC stays all-1s
    const int halfsel = lane >> 4;                  // 0 or 1
    const int l16     = lane & 15;

    v8f acc[4] = {};

#if HAVE_WMMA_F32
    const int koff = halfsel * 2;
    const float* __restrict__ xr = X + (size_t)(tile * 16 + l16) * IN_F;
    for (int k0 = 0; k0 < IN_F; k0 += 4) {
        v2f a;
        a.x = xr[k0 + koff];
        a.y = xr[k0 + koff + 1];
        const float* __restrict__ wp = W1 + (size_t)(k0 + koff) * H_F + l16;
#pragma unroll
        for (int n = 0; n < 4; ++n) {
            v2f b;
            b.x = wp[n * 16];          // row k0+koff,   col n*16+l16
            b.y = wp[n * 16 + H_F];    // row k0+koff+1, col n*16+l16
            acc[n] = __builtin_amdgcn_wmma_f32_16x16x4_f32(
                false, a, false, b, (short)0, acc[n], false, false);
        }
    }
#else
    for (int n = 0; n < 4; ++n)
        for (int v = 0; v < 8; ++v) {
            const int row = tile * 16 + v + 8 * halfsel;
            const int col = n * 16 + l16;
            float s = 0.f;
            for (int k = 0; k < IN_F; ++k)
                s = fmaf(X[(size_t)row * IN_F + k], W1[(size_t)k * H_F + col], s);
            acc[n][v] = s;
        }
#endif

#pragma unroll
    for (int n = 0; n < 4; ++n)
#pragma unroll
        for (int v = 0; v < 8; ++v)
            T1[(size_t)(tile * 16 + v + 8 * halfsel) * H_F + n * 16 + l16] = acc[n][v];
}

// ---------------------------------------------------------------------------
// GEMM2: T2[M,16] = relu(AGG1[M,64] + b1) @ W2[64,16]
// relu+b1 fused into the A-fragment load; all 16 B fragments (32 VGPRs) and
// the b1 pairs hoisted out of the K loop.
// ---------------------------------------------------------------------------
__global__ __launch_bounds__(256) void gcn_gemm2(
    const float* __restrict__ AGG1, const float* __restrict__ b1,
    const float* __restrict__ W2, float* __restrict__ T2, int n_tiles)
{
    const int wave    = threadIdx.x >> 5;
    const int lane    = threadIdx.x & 31;
    const int tile    = blockIdx.x * 8 + wave;
    if (tile >= n_tiles) return;
    const int halfsel = lane >> 4;
    const int l16     = lane & 15;

    v8f acc = {};

#if HAVE_WMMA_F32
    const int koff = halfsel * 2;

    v2f bfrag[16];
    v2f b1v[16];
#pragma unroll
    for (int i = 0; i < 16; ++i) {
        const int ka = i * 4 + koff;
        bfrag[i].x = W2[(size_t)ka       * C_F + l16];
        bfrag[i].y = W2[(size_t)(ka + 1) * C_F + l16];
        b1v[i].x   = b1[ka];
        b1v[i].y   = b1[ka + 1];
    }

    const float* __restrict__ ar = AGG1 + (size_t)(tile * 16 + l16) * H_F;
#pragma unroll
    for (int i = 0; i < 16; ++i) {
        const int ka = i * 4 + koff;
        v2f a;
        a.x = fmaxf(ar[ka]     + b1v[i].x, 0.f);
        a.y = fmaxf(ar[ka + 1] + b1v[i].y, 0.f);
        acc = __builtin_amdgcn_wmma_f32_16x16x4_f32(
            false, a, false, bfrag[i], (short)0, acc, false, false);
    }
#else
    for (int v = 0; v < 8; ++v) {
        const int row = tile * 16 + v + 8 * halfsel;
        float s = 0.f;
        for (int k = 0; k < H_F; ++k) {
            const float h = fmaxf(AGG1[(size_t)row * H_F + k] + b1[k], 0.f);
            s = fmaf(h, W2[(size_t)k * C_F + l16], s);
        }
        acc[v] = s;
    }
#endif

#pragma unroll
    for (int v = 0; v < 8; ++v)
        T2[(size_t)(tile * 16 + v + 8 * halfsel) * C_F + l16] = acc[v];
}

// ---------------------------------------------------------------------------
// Scatter 1: one wave32 per edge, 64 features (2 per lane).
// Edge index forced wave-uniform via readfirstlane so src/dst/weight become
// scalar (SMEM) loads; gathers are 128B-coalesced; native f32 atomics out.
// ---------------------------------------------------------------------------
__global__ __launch_bounds__(256) void gcn_scatter64(
    const float* __restrict__ T1, const int* __restrict__ esrc,
    const int* __restrict__ edst, const float* __restrict__ ew,
    float* __restrict__ AGG1, int n_edges)
{
    const int e = __builtin_amdgcn_readfirstlane(
        blockIdx.x * 8 + (threadIdx.x >> 5));       // wave-uniform
    const int lane = threadIdx.x & 31;
    if (e >= n_edges) return;
    const int   s = esrc[e];
    const int   d = edst[e];
    const float w = ew[e];
    const float* __restrict__ ts = T1 + (size_t)s * H_F;
    float* __restrict__       ad = AGG1 + (size_t)d * H_F;
    const float v0 = ts[lane]      * w;
    const float v1 = ts[lane + 32] * w;
    atomic_add_f32_native(&ad[lane],      v0);
    atomic_add_f32_native(&ad[lane + 32], v1);
}

// ---------------------------------------------------------------------------
// Scatter 2: 16 lanes per edge (one per class) into d_out (pre-seeded with b2).
// ---------------------------------------------------------------------------
__global__ __launch_bounds__(256) void gcn_scatter16(
    const float* __restrict__ T2, const int* __restrict__ esrc,
    const int* __restrict__ edst, const float* __restrict__ ew,
    float* __restrict__ OUT, int n_edges)
{
    const int gid = blockIdx.x * 256 + threadIdx.x;
    const int e   = gid >> 4;
    const int f   = gid & 15;
    if (e >= n_edges) return;
    const float v = T2[(size_t)esrc[e] * C_F + f] * ew[e];
    atomic_add_f32_native(&OUT[(size_t)edst[e] * C_F + f], v);
}

__global__ __launch_bounds__(256) void gcn_zero(float* __restrict__ p, int n)
{
    const int i = blockIdx.x * 256 + threadIdx.x;
    if (i < n) p[i] = 0.f;
}

__global__ __launch_bounds__(256) void gcn_seed_bias(
    float* __restrict__ out, const float* __restrict__ b2, int n)
{
    const int i = blockIdx.x * 256 + threadIdx.x;
    if (i < n) out[i] = b2[i & (C_F - 1)];
}

// ---------------------------------------------------------------------------
extern "C" void kernel_launch(void* const* d_in, const int* in_sizes, int n_in,
                              void* d_out, int out_size, void* d_ws, size_t ws_size,
                              hipStream_t stream)
{
    const float* X    = (const float*)d_in[0];
    const float* ew   = (const float*)d_in[1];
    const float* W1   = (const float*)d_in[2];
    const float* b1   = (const float*)d_in[3];
    const float* W2   = (const float*)d_in[4];
    const float* b2   = (const float*)d_in[5];
    const int*   esrc = (const int*)d_in[6];
    const int*   edst = (const int*)d_in[7];
    float*       out  = (float*)d_out;

    const int n_nodes = in_sizes[0] / IN_F;   // 100000
    const int n_edges = in_sizes[1];          // 1600000
    const int n_tiles = n_nodes / 16;         // 6250 (exact)

    // workspace partition: T1 (N*64) | agg1 (N*64) | T2 (N*16)  ~= 57.6 MB
    float* T1   = (float*)d_ws;
    float* agg1 = T1   + (size_t)n_nodes * H_F;
    float* T2   = agg1 + (size_t)n_nodes * H_F;

    const int gemm_blocks = (n_tiles + 7) / 8;

    // Layer 1
    gcn_gemm1<<<gemm_blocks, 256, 0, stream>>>(X, W1, T1, n_tiles);
    const int agg1_n = n_nodes * H_F;
    gcn_zero<<<(agg1_n + 255) / 256, 256, 0, stream>>>(agg1, agg1_n);
    gcn_scatter64<<<(n_edges + 7) / 8, 256, 0, stream>>>(T1, esrc, edst, ew, agg1, n_edges);

    // Layer 2 (relu + b1 fused into GEMM2 A-fragment load)
    gcn_gemm2<<<gemm_blocks, 256, 0, stream>>>(agg1, b1, W2, T2, n_tiles);
    gcn_seed_bias<<<(out_size + 255) / 256, 256, 0, stream>>>(out, b2, out_size);
    gcn_scatter16<<<((size_t)n_edges * C_F + 255) / 256, 256, 0, stream>>>(
        T2, esrc, edst, ew, out, n_edges);
}